// SECONDBackbone_8143257993860
// MI455X (gfx1250) — compile-verified
//
#include <hip/hip_runtime.h>
#include <math.h>

typedef float v2f __attribute__((ext_vector_type(2)));
typedef float v8f __attribute__((ext_vector_type(8)));

#define BB 2
#define CC 128
#define HH 256
#define WW 256
#define NCELLS (BB * HH * WW)          /* 131072 */
#define CPB (HH * WW)                  /* 65536 cells per batch */
#define TOTAL ((size_t)BB * CC * HH * WW) /* 16777216 */

// ---------------- kernel 1: zero winner keys ----------------
__global__ void k_init_keys(unsigned long long* __restrict__ keys) {
  int i = blockIdx.x * blockDim.x + threadIdx.x;
  if (i < NCELLS) keys[i] = 0ull;
}

// ---------------- kernel 2: scatter-max packed (normbits, voxid+1) ----------
// one wave per voxel: coalesced 512B read, cross-lane reduce, u64 atomic max
__global__ void k_scatter(const float* __restrict__ features,
                          const int* __restrict__ batch_idx,
                          const int* __restrict__ w_idx,
                          const int* __restrict__ d_idx,
                          unsigned long long* __restrict__ keys, int n) {
  int gtid = blockIdx.x * blockDim.x + threadIdx.x;
  int v = gtid >> 5;       // uniform within a wave
  int lane = gtid & 31;
  if (v >= n) return;      // whole wave exits together
  const float4* f4 = (const float4*)(features + (size_t)v * CC);
  float4 x = f4[lane];
  float s = fabsf(x.x) + fabsf(x.y) + fabsf(x.z) + fabsf(x.w);
#pragma unroll
  for (int m = 16; m > 0; m >>= 1) s += __shfl_xor(s, m, 32);
  if (lane == 0) {
    int b = batch_idx[v], h = w_idx[v], w = d_idx[v];
    int cell = (b * HH + h) * WW + w;
    unsigned long long key =
        ((unsigned long long)__float_as_uint(s) << 32) | (unsigned)(v + 1);
    atomicMax(&keys[cell], key);
  }
}

// ---------------- kernel 3: gather + LDS transpose + WMMA channel sums ------
// block = 256 threads = one BEV row (256 cells, same b,h).
// WMMA f32 16x16x4 with ones-A: D[m][n] = sum_k B[k][n]  -> column sums.
// Lane l places values for channel c0 + (l%16); D vgpr0 lane l returns the
// column sum for that same channel (N = lane%16 for B and D halves alike).
__global__ void k_gather(const float* __restrict__ features,
                         const unsigned long long* __restrict__ keys,
                         float* __restrict__ out,
                         float* __restrict__ part_sum,
                         float* __restrict__ part_sq) {
  __shared__ float tile[64 * 129];  // 64 cells x 128 ch, padded stride
  __shared__ int vox_s[256];
  const int row = blockIdx.x;  // b*256 + h
  const int b = row >> 8;
  const int h = row & 255;
  const int t = threadIdx.x;
  const int lane = t & 31;
  const int wave = t >> 5;
  const int c0 = wave << 4;        // 16 channels per wave
  const int nsub = lane & 15;
  const int half = lane >> 4;

  {
    unsigned long long key = keys[(size_t)row * 256 + t];
    vox_s[t] = key ? (int)(unsigned)(key & 0xffffffffull) - 1 : -1;
  }

  v2f aones;
  aones[0] = 1.0f;
  aones[1] = 1.0f;
  v8f acc, accq;
#pragma unroll
  for (int q = 0; q < 8; ++q) {
    acc[q] = 0.0f;
    accq[q] = 0.0f;
  }

  const int myc = c0 + nsub;
  for (int chunk = 0; chunk < 4; ++chunk) {
    __syncthreads();
    // gather winner features, cell-major (coalesced 512B runs per voxel)
    for (int i = t; i < 64 * 128; i += 256) {
      int j = i >> 7, ch = i & 127;
      int vox = vox_s[(chunk << 6) + j];
      float val = (vox >= 0) ? features[(size_t)vox * CC + ch] : 0.0f;
      tile[j * 129 + ch] = val;
    }
    __syncthreads();
    // WMMA column-sum accumulate: 4 cells per issue, 16 channels per wave
#pragma unroll
    for (int g = 0; g < 16; ++g) {
      int j0 = (g << 2) + half;  // cells j0 and j0+2
      float v0 = tile[j0 * 129 + myc];
      float v1 = tile[(j0 + 2) * 129 + myc];
      v2f bv, bq;
      bv[0] = v0;
      bv[1] = v1;
      bq[0] = v0 * v0;
      bq[1] = v1 * v1;
      acc = __builtin_amdgcn_wmma_f32_16x16x4_f32(false, aones, false, bv,
                                                  (short)0, acc, false, false);
      accq = __builtin_amdgcn_wmma_f32_16x16x4_f32(false, aones, false, bq,
                                                   (short)0, accq, false, false);
    }
    // write raw BEV channel-major (coalesced 256B runs)
    for (int i = t; i < 64 * 128; i += 256) {
      int ch = i >> 6, j = i & 63;
      out[(((size_t)b * CC + ch) << 16) + ((size_t)h << 8) + (chunk << 6) + j] =
          tile[j * 129 + ch];
    }
  }
  // every lane's acc[0] holds colsum(lane%16); take lanes 0..15 once
  if (lane < 16) {
    part_sum[(size_t)row * CC + c0 + lane] = acc[0];
    part_sq[(size_t)row * CC + c0 + lane] = accq[0];
  }
}

// ---------------- kernel 4: deterministic stats (softmax + layernorm) -------
__global__ void k_stats(const float* __restrict__ part_sum,
                        const float* __restrict__ part_sq,
                        float* __restrict__ wbuf,
                        float* __restrict__ stats) {
  __shared__ float red[128];
  const int b = blockIdx.x;
  const int c = threadIdx.x;  // 128 threads = channels
  float s = 0.0f, sq = 0.0f;
  for (int r = 0; r < 256; ++r) {
    size_t off = (size_t)(b * 256 + r) * CC + c;
    s += part_sum[off];
    sq += part_sq[off];
  }
  float chmean = s * (1.0f / (float)CPB);
  // softmax over channels: max
  red[c] = chmean;
  __syncthreads();
  for (int st = 64; st > 0; st >>= 1) {
    if (c < st) red[c] = fmaxf(red[c], red[c + st]);
    __syncthreads();
  }
  float mx = red[0];
  __syncthreads();
  float e = expf(chmean - mx);
  red[c] = e;
  __syncthreads();
  for (int st = 64; st > 0; st >>= 1) {
    if (c < st) red[c] += red[c + st];
    __syncthreads();
  }
  float se = red[0];
  __syncthreads();
  float w = e / se;
  // mean of (bev*w) over (C,H,W):  sum_c w_c * sum_hw
  red[c] = w * s;
  __syncthreads();
  for (int st = 64; st > 0; st >>= 1) {
    if (c < st) red[c] += red[c + st];
    __syncthreads();
  }
  float Sw = red[0];
  __syncthreads();
  // E[x^2]: sum_c w_c^2 * sumsq_hw
  red[c] = w * w * sq;
  __syncthreads();
  for (int st = 64; st > 0; st >>= 1) {
    if (c < st) red[c] += red[c + st];
    __syncthreads();
  }
  float Sq = red[0];
  __syncthreads();
  const float inv = 1.0f / (float)((size_t)CC * CPB);
  float mean = Sw * inv;
  float var = Sq * inv - mean * mean;
  float rstd = rsqrtf(var + 1e-5f);
  wbuf[b * CC + c] = w;
  if (c == 0) {
    stats[b * 2] = mean;
    stats[b * 2 + 1] = rstd;
  }
}

// ---------------- kernel 5: in-place reweight + normalize + mask ------------
__global__ void k_finalize(float* __restrict__ out,
                           const unsigned* __restrict__ keys_u32,
                           const float* __restrict__ wbuf,
                           const float* __restrict__ stats) {
  size_t i = (size_t)blockIdx.x * blockDim.x + threadIdx.x;
  int b = (int)(i >> 23);              // C*H*W = 2^23
  int c = (int)((i >> 16) & 127);
  int cell = (int)(i & 65535);
  unsigned hi = keys_u32[2 * (((size_t)b << 16) + cell) + 1];  // norm bits
  float raw = out[i];
  float val = raw * wbuf[b * CC + c];
  float mean = stats[b * 2];
  float rstd = stats[b * 2 + 1];
  out[i] = (hi != 0u) ? (val - mean) * rstd : 0.0f;
}

extern "C" void kernel_launch(void* const* d_in, const int* in_sizes, int n_in,
                              void* d_out, int out_size, void* d_ws,
                              size_t ws_size, hipStream_t stream) {
  const float* features = (const float*)d_in[0];
  const int* batch_idx = (const int*)d_in[1];
  const int* w_idx = (const int*)d_in[2];
  const int* d_idx = (const int*)d_in[3];
  const int n = in_sizes[1];  // number of voxels
  float* out = (float*)d_out;

  char* ws = (char*)d_ws;
  unsigned long long* keys = (unsigned long long*)ws;          // 1 MB
  float* part_sum = (float*)(ws + 1048576);                    // 256 KB
  float* part_sq = (float*)(ws + 1048576 + 262144);            // 256 KB
  float* wbuf = (float*)(ws + 1048576 + 524288);               // 1 KB
  float* stats = (float*)(ws + 1048576 + 524288 + 1024);       // 16 B
  (void)ws_size;
  (void)n_in;
  (void)out_size;

  k_init_keys<<<NCELLS / 256, 256, 0, stream>>>(keys);
  k_scatter<<<(n + 7) / 8, 256, 0, stream>>>(features, batch_idx, w_idx, d_idx,
                                             keys, n);
  k_gather<<<BB * HH, 256, 0, stream>>>(features, keys, out, part_sum, part_sq);
  k_stats<<<BB, CC, 0, stream>>>(part_sum, part_sq, wbuf, stats);
  k_finalize<<<(unsigned)(TOTAL / 256), 256, 0, stream>>>(
      out, (const unsigned*)keys, wbuf, stats);
}